// S5_29274497090136
// MI455X (gfx1250) — compile-verified
//
#include <hip/hip_runtime.h>
#include <cmath>

// Problem constants (match reference)
#define L_SEQ 8192
#define H_DIM 1024
#define P_DIM 256
#define SCHUNK 64
#define NCHUNK (L_SEQ / SCHUNK)   // 128

typedef float v2f __attribute__((ext_vector_type(2)));
typedef float v8f __attribute__((ext_vector_type(8)));

// D = A(16x4, f32) * B(4x16, f32) + C(16x16, f32)
__device__ __forceinline__ v8f wmma4(v2f a, v2f b, v8f c) {
  return __builtin_amdgcn_wmma_f32_16x16x4_f32(
      /*neg_a=*/false, a, /*neg_b=*/false, b,
      /*c_mod=*/(short)0, c, /*reuse_a=*/false, /*reuse_b=*/false);
}

// CDNA5 async global->LDS copies (ASYNCcnt tracked), bypass VGPR staging.
__device__ __forceinline__ void async_ld_b128(unsigned lds_off, const float* g) {
  asm volatile("global_load_async_to_lds_b128 %0, %1, off"
               :: "v"(lds_off), "v"((unsigned long long)(size_t)g) : "memory");
}
__device__ __forceinline__ void async_ld_b64(unsigned lds_off, const float* g) {
  asm volatile("global_load_async_to_lds_b64 %0, %1, off"
               :: "v"(lds_off), "v"((unsigned long long)(size_t)g) : "memory");
}
__device__ __forceinline__ void wait_async0() {
  asm volatile("s_wait_asynccnt 0x0" ::: "memory");
}
__device__ __forceinline__ unsigned lds_off(const void* p) {
  return (unsigned)(size_t)p;   // low 32 bits of generic addr = LDS byte offset
}

// ---------------------------------------------------------------------------
// Kernel 0: per-channel discretization.  Lambda_bar = exp(Lam*step),
// coef = (Lambda_bar - 1)/Lam, LamS = Lambda_bar^SCHUNK = exp(Lam*step*S).
// ---------------------------------------------------------------------------
__global__ void k_setup(const float* __restrict__ Lre, const float* __restrict__ Lim,
                        const float* __restrict__ logstep,
                        float* __restrict__ lam_re, float* __restrict__ lam_im,
                        float* __restrict__ lamS_re, float* __restrict__ lamS_im,
                        float* __restrict__ coef_re, float* __restrict__ coef_im) {
  const int p = threadIdx.x;
  const float lr = Lre[p], li = Lim[p];
  const float st = expf(logstep[p]);
  const float a = lr * st, b = li * st;
  const float ea = expf(a);
  const float er = ea * cosf(b), ei = ea * sinf(b);   // Lambda_bar
  lam_re[p] = er; lam_im[p] = ei;
  const float aS = a * (float)SCHUNK, bS = b * (float)SCHUNK;
  const float eaS = expf(aS);
  lamS_re[p] = eaS * cosf(bS);
  lamS_im[p] = eaS * sinf(bS);
  const float d  = lr * lr + li * li;
  const float nr = er - 1.0f;
  coef_re[p] = (nr * lr + ei * li) / d;               // (Lbar-1)/Lam
  coef_im[p] = (ei * lr - nr * li) / d;
}

// ---------------------------------------------------------------------------
// Kernel 0b: B_barT[h*P+p] = coef[p] * B_tilde[p,h]   (transposed to (H,P))
// ---------------------------------------------------------------------------
__global__ void k_prep_B(const float* __restrict__ B,
                         const float* __restrict__ coef_re, const float* __restrict__ coef_im,
                         float* __restrict__ BbT_re, float* __restrict__ BbT_im) {
  const int idx = blockIdx.x * 256 + threadIdx.x;     // idx = h*P + p
  const int h = idx >> 8;
  const int p = idx & (P_DIM - 1);
  const float cr = coef_re[p], ci = coef_im[p];
  const float br = B[((size_t)p * H_DIM + h) * 2 + 0];
  const float bi = B[((size_t)p * H_DIM + h) * 2 + 1];
  BbT_re[idx] = cr * br - ci * bi;
  BbT_im[idx] = cr * bi + ci * br;
}

// ---------------------------------------------------------------------------
// Kernel 0c: CrT[p*H+h] = 2*C_re[h,p];  CiT[p*H+h] = -2*C_im[h,p]
// ---------------------------------------------------------------------------
__global__ void k_prep_C(const float* __restrict__ C,
                         float* __restrict__ CrT, float* __restrict__ CiT) {
  const int idx = blockIdx.x * 256 + threadIdx.x;     // idx = p*H + h
  const int p = idx >> 10;
  const int h = idx & (H_DIM - 1);
  CrT[idx] =  2.0f * C[((size_t)h * P_DIM + p) * 2 + 0];
  CiT[idx] = -2.0f * C[((size_t)h * P_DIM + p) * 2 + 1];
}

// LDS tile pitches (floats).
// A pitch 20: b128 dests 16B aligned; 20*r mod 64 distinct for r in [0,16),
//             and 20*dr == 2 (mod 64) unsolvable -> conflict-free frag reads.
// B pitch 40: b64 dests 8B aligned; lane-half row delta 2*40 == 16 (mod 64)
//             -> halves occupy disjoint bank ranges, conflict-free.
#define APITCH 20
#define BPITCH 40

// ---------------------------------------------------------------------------
// Kernel 1: Bu(L,P) = U(L,H) @ B_barT(H,P), complex result (re,im).
// 256 thr = 8 waves (4M x 2N); block tile 64x32; K chunks of 16 staged in
// double-buffered LDS via async global->LDS copies.
// ---------------------------------------------------------------------------
__global__ void __launch_bounds__(256) k_gemm_bu(
    const float* __restrict__ U,
    const float* __restrict__ BbT_re, const float* __restrict__ BbT_im,
    float* __restrict__ Bu_re, float* __restrict__ Bu_im) {
  __shared__ __align__(16) float As[2][64][APITCH];
  __shared__ __align__(16) float Bsr[2][16][BPITCH];   // [k][n]
  __shared__ __align__(16) float Bsi[2][16][BPITCH];

  const int tid  = threadIdx.x;
  const int lane = tid & 31, wave = tid >> 5;
  const int wm = wave & 3, wn = wave >> 2;
  const int l15 = lane & 15, lh = lane >> 4;
  const int p0 = blockIdx.x * 32;
  const int l0 = blockIdx.y * 64;

  v8f accr = {}; v8f acci = {};

  const int arow = tid >> 2, acol = (tid & 3) * 4;    // A tile: 64 x 16
  const int bk = tid >> 4,  bn = (tid & 15) * 2;      // B tile: 16 x 32

  const float* gA  = U + (size_t)(l0 + arow) * H_DIM + acol;
  const float* gBr = BbT_re + (size_t)bk * P_DIM + p0 + bn;
  const float* gBi = BbT_im + (size_t)bk * P_DIM + p0 + bn;

  // prologue: chunk 0 into buffer 0
  async_ld_b128(lds_off(&As[0][arow][acol]), gA);
  async_ld_b64(lds_off(&Bsr[0][bk][bn]), gBr);
  async_ld_b64(lds_off(&Bsi[0][bk][bn]), gBi);
  wait_async0();
  __syncthreads();

  const int NKC = H_DIM / 16;   // 64
  for (int ci = 0; ci < NKC; ++ci) {
    const int cur = ci & 1;
    if (ci + 1 < NKC) {
      const int nxt = cur ^ 1;
      const int h1 = (ci + 1) * 16;
      async_ld_b128(lds_off(&As[nxt][arow][acol]), gA + h1);
      async_ld_b64(lds_off(&Bsr[nxt][bk][bn]), gBr + (size_t)h1 * P_DIM);
      async_ld_b64(lds_off(&Bsi[nxt][bk][bn]), gBi + (size_t)h1 * P_DIM);
    }
#pragma unroll
    for (int kk = 0; kk < 16; kk += 4) {
      const int ka = kk + lh * 2;
      v2f a, br, bi;
      a[0]  = As[cur][wm * 16 + l15][ka + 0];
      a[1]  = As[cur][wm * 16 + l15][ka + 1];
      br[0] = Bsr[cur][ka + 0][wn * 16 + l15];
      br[1] = Bsr[cur][ka + 1][wn * 16 + l15];
      bi[0] = Bsi[cur][ka + 0][wn * 16 + l15];
      bi[1] = Bsi[cur][ka + 1][wn * 16 + l15];
      accr = wmma4(a, br, accr);
      acci = wmma4(a, bi, acci);
    }
    wait_async0();
    __syncthreads();
  }

  const int orow0 = l0 + wm * 16 + 8 * lh;            // M = v + 8*(lane>=16)
  const int ocol  = p0 + wn * 16 + l15;               // N = lane & 15
#pragma unroll
  for (int v = 0; v < 8; ++v) {
    Bu_re[(size_t)(orow0 + v) * P_DIM + ocol] = accr[v];
    Bu_im[(size_t)(orow0 + v) * P_DIM + ocol] = acci[v];
  }
}

// ---------------------------------------------------------------------------
// Kernel 2a: per-chunk local scan with zero init; store only chunk finals.
// ---------------------------------------------------------------------------
__global__ void k_scan_local(const float* __restrict__ Bu_re, const float* __restrict__ Bu_im,
                             const float* __restrict__ lam_re, const float* __restrict__ lam_im,
                             float* __restrict__ fin_re, float* __restrict__ fin_im) {
  const int p = threadIdx.x;
  const int c = blockIdx.x;
  const float lr = lam_re[p], li = lam_im[p];
  float xr = 0.f, xi = 0.f;
  size_t base = (size_t)c * SCHUNK * P_DIM + p;
  for (int j = 0; j < SCHUNK; ++j) {
    const float br = Bu_re[base], bi = Bu_im[base];
    const float nxr = fmaf(lr, xr, fmaf(-li, xi, br));
    const float nxi = fmaf(lr, xi, fmaf( li, xr, bi));
    xr = nxr; xi = nxi;
    base += P_DIM;
  }
  fin_re[(size_t)c * P_DIM + p] = xr;
  fin_im[(size_t)c * P_DIM + p] = xi;
}

// ---------------------------------------------------------------------------
// Kernel 2b: exclusive scan over chunk finals with constant factor Lam^S.
// ---------------------------------------------------------------------------
__global__ void k_scan_carry(const float* __restrict__ fin_re, const float* __restrict__ fin_im,
                             const float* __restrict__ lamS_re, const float* __restrict__ lamS_im,
                             float* __restrict__ car_re, float* __restrict__ car_im) {
  const int p = threadIdx.x;
  const float lr = lamS_re[p], li = lamS_im[p];
  float sr = 0.f, si = 0.f;
  for (int c = 0; c < NCHUNK; ++c) {
    car_re[(size_t)c * P_DIM + p] = sr;
    car_im[(size_t)c * P_DIM + p] = si;
    const float fr = fin_re[(size_t)c * P_DIM + p];
    const float fi = fin_im[(size_t)c * P_DIM + p];
    const float nsr = fmaf(lr, sr, fmaf(-li, si, fr));
    const float nsi = fmaf(lr, si, fmaf( li, sr, fi));
    sr = nsr; si = nsi;
  }
}

// ---------------------------------------------------------------------------
// Kernel 2c: re-run recurrence seeded with carry; overwrite Bu -> xs in place.
// ---------------------------------------------------------------------------
__global__ void k_scan_apply(float* __restrict__ Bu_re, float* __restrict__ Bu_im,
                             const float* __restrict__ lam_re, const float* __restrict__ lam_im,
                             const float* __restrict__ car_re, const float* __restrict__ car_im) {
  const int p = threadIdx.x;
  const int c = blockIdx.x;
  const float lr = lam_re[p], li = lam_im[p];
  float xr = car_re[(size_t)c * P_DIM + p];
  float xi = car_im[(size_t)c * P_DIM + p];
  size_t base = (size_t)c * SCHUNK * P_DIM + p;
  for (int j = 0; j < SCHUNK; ++j) {
    const float br = Bu_re[base], bi = Bu_im[base];
    const float nxr = fmaf(lr, xr, fmaf(-li, xi, br));
    const float nxi = fmaf(lr, xi, fmaf( li, xr, bi));
    xr = nxr; xi = nxi;
    Bu_re[base] = xr; Bu_im[base] = xi;
    base += P_DIM;
  }
}

// ---------------------------------------------------------------------------
// Kernel 3: Out(L,H) = Xr @ (2CrT) + Xi @ (-2CiT) + D .* U
// Same double-buffered async-LDS structure; K = P_DIM.
// ---------------------------------------------------------------------------
__global__ void __launch_bounds__(256) k_gemm_out(
    const float* __restrict__ Xr, const float* __restrict__ Xi,
    const float* __restrict__ CrT, const float* __restrict__ CiT,
    const float* __restrict__ U, const float* __restrict__ Dvec,
    float* __restrict__ Out) {
  __shared__ __align__(16) float Ar[2][64][APITCH];
  __shared__ __align__(16) float Ai[2][64][APITCH];
  __shared__ __align__(16) float Bsr[2][16][BPITCH];   // [k][n]
  __shared__ __align__(16) float Bsi[2][16][BPITCH];

  const int tid  = threadIdx.x;
  const int lane = tid & 31, wave = tid >> 5;
  const int wm = wave & 3, wn = wave >> 2;
  const int l15 = lane & 15, lh = lane >> 4;
  const int h0 = blockIdx.x * 32;
  const int l0 = blockIdx.y * 64;

  v8f acc = {};

  const int arow = tid >> 2, acol = (tid & 3) * 4;
  const int bk = tid >> 4,  bn = (tid & 15) * 2;

  const float* gAr = Xr + (size_t)(l0 + arow) * P_DIM + acol;
  const float* gAi = Xi + (size_t)(l0 + arow) * P_DIM + acol;
  const float* gBr = CrT + (size_t)bk * H_DIM + h0 + bn;
  const float* gBi = CiT + (size_t)bk * H_DIM + h0 + bn;

  async_ld_b128(lds_off(&Ar[0][arow][acol]), gAr);
  async_ld_b128(lds_off(&Ai[0][arow][acol]), gAi);
  async_ld_b64(lds_off(&Bsr[0][bk][bn]), gBr);
  async_ld_b64(lds_off(&Bsi[0][bk][bn]), gBi);
  wait_async0();
  __syncthreads();

  const int NKC = P_DIM / 16;   // 16
  for (int ci = 0; ci < NKC; ++ci) {
    const int cur = ci & 1;
    if (ci + 1 < NKC) {
      const int nxt = cur ^ 1;
      const int k1 = (ci + 1) * 16;
      async_ld_b128(lds_off(&Ar[nxt][arow][acol]), gAr + k1);
      async_ld_b128(lds_off(&Ai[nxt][arow][acol]), gAi + k1);
      async_ld_b64(lds_off(&Bsr[nxt][bk][bn]), gBr + (size_t)k1 * H_DIM);
      async_ld_b64(lds_off(&Bsi[nxt][bk][bn]), gBi + (size_t)k1 * H_DIM);
    }
#pragma unroll
    for (int kk = 0; kk < 16; kk += 4) {
      const int ka = kk + lh * 2;
      v2f ar, ai, br, bi;
      ar[0] = Ar[cur][wm * 16 + l15][ka + 0];
      ar[1] = Ar[cur][wm * 16 + l15][ka + 1];
      ai[0] = Ai[cur][wm * 16 + l15][ka + 0];
      ai[1] = Ai[cur][wm * 16 + l15][ka + 1];
      br[0] = Bsr[cur][ka + 0][wn * 16 + l15];
      br[1] = Bsr[cur][ka + 1][wn * 16 + l15];
      bi[0] = Bsi[cur][ka + 0][wn * 16 + l15];
      bi[1] = Bsi[cur][ka + 1][wn * 16 + l15];
      acc = wmma4(ar, br, acc);
      acc = wmma4(ai, bi, acc);
    }
    wait_async0();
    __syncthreads();
  }

  const int orow0 = l0 + wm * 16 + 8 * lh;
  const int ocol  = h0 + wn * 16 + l15;
  const float dv = Dvec[ocol];
#pragma unroll
  for (int v = 0; v < 8; ++v) {
    const size_t o = (size_t)(orow0 + v) * H_DIM + ocol;
    Out[o] = acc[v] + dv * U[o];
  }
}

// ---------------------------------------------------------------------------
extern "C" void kernel_launch(void* const* d_in, const int* in_sizes, int n_in,
                              void* d_out, int out_size, void* d_ws, size_t ws_size,
                              hipStream_t stream) {
  (void)in_sizes; (void)n_in; (void)out_size; (void)ws_size;

  const float* U     = (const float*)d_in[0];   // (L,H)
  const float* Lre   = (const float*)d_in[1];   // (P,)
  const float* Lim   = (const float*)d_in[2];   // (P,)
  const float* B     = (const float*)d_in[3];   // (P,H,2)
  const float* C     = (const float*)d_in[4];   // (H,P,2)
  const float* Dv    = (const float*)d_in[5];   // (H,)
  const float* lstep = (const float*)d_in[6];   // (P,)
  float* Out = (float*)d_out;                   // (L,H)

  float* ws = (float*)d_ws;
  const size_t HP = (size_t)H_DIM * P_DIM;
  const size_t CP = (size_t)NCHUNK * P_DIM;
  const size_t LP = (size_t)L_SEQ * P_DIM;

  float* BbT_re  = ws;                 // (H,P)
  float* BbT_im  = BbT_re + HP;        // (H,P)
  float* CrT     = BbT_im + HP;        // (P,H)  = 2*C_re^T
  float* CiT     = CrT + HP;           // (P,H)  = -2*C_im^T
  float* lam_re  = CiT + HP;           // (P,)
  float* lam_im  = lam_re + P_DIM;
  float* lamS_re = lam_im + P_DIM;
  float* lamS_im = lamS_re + P_DIM;
  float* coef_re = lamS_im + P_DIM;
  float* coef_im = coef_re + P_DIM;
  float* fin_re  = coef_im + P_DIM;    // (NCHUNK,P)
  float* fin_im  = fin_re + CP;
  float* car_re  = fin_im + CP;        // (NCHUNK,P)
  float* car_im  = car_re + CP;
  float* Bu_re   = car_im + CP;        // (L,P), becomes xs in place
  float* Bu_im   = Bu_re + LP;

  k_setup<<<1, P_DIM, 0, stream>>>(Lre, Lim, lstep,
                                   lam_re, lam_im, lamS_re, lamS_im, coef_re, coef_im);
  k_prep_B<<<(int)(HP / 256), 256, 0, stream>>>(B, coef_re, coef_im, BbT_re, BbT_im);
  k_prep_C<<<(int)(HP / 256), 256, 0, stream>>>(C, CrT, CiT);
  k_gemm_bu<<<dim3(P_DIM / 32, L_SEQ / 64), 256, 0, stream>>>(U, BbT_re, BbT_im, Bu_re, Bu_im);
  k_scan_local<<<NCHUNK, P_DIM, 0, stream>>>(Bu_re, Bu_im, lam_re, lam_im, fin_re, fin_im);
  k_scan_carry<<<1, P_DIM, 0, stream>>>(fin_re, fin_im, lamS_re, lamS_im, car_re, car_im);
  k_scan_apply<<<NCHUNK, P_DIM, 0, stream>>>(Bu_re, Bu_im, lam_re, lam_im, car_re, car_im);
  k_gemm_out<<<dim3(H_DIM / 32, L_SEQ / 64), 256, 0, stream>>>(Bu_re, Bu_im, CrT, CiT, U, Dv, Out);
}